// SAGE_28192165331560
// MI455X (gfx1250) — compile-verified
//
#include <hip/hip_runtime.h>
#include <hip/hip_bf16.h>
#include <math.h>

// ---------------- problem constants (match reference) ----------------
#define NN    50000
#define EE    800000
#define IN_C  128
#define HID_C 256
#define OUT_C 128
#define BN_EPS 1e-5f

typedef __attribute__((ext_vector_type(2))) float v2f;
typedef __attribute__((ext_vector_type(8))) float v8f;

// ---------------------------------------------------------------------
// Edge-count (mean denominator) via f32 atomics; inverted in place.
// ---------------------------------------------------------------------
__global__ void count_kernel(const long long* __restrict__ dst,
                             float* __restrict__ cnt, int E) {
    int e = blockIdx.x * blockDim.x + threadIdx.x;
    if (e < E) atomicAdd(&cnt[(int)dst[e]], 1.0f);
}

__global__ void inv_kernel(float* __restrict__ cnt, int n) {
    int i = blockIdx.x * blockDim.x + threadIdx.x;
    if (i < n) cnt[i] = 1.0f / fmaxf(cnt[i], 1.0f);
}

// ---------------------------------------------------------------------
// Scatter-add: one block per edge, one channel per thread (blockDim == F).
// Feature rows are 512B/1KB contiguous -> coalesced; accumulator is
// L2-resident (51MB << 192MB L2) so atomics run at L2 rates.
// ---------------------------------------------------------------------
__global__ void scatter_add_kernel(const long long* __restrict__ src,
                                   const long long* __restrict__ dst,
                                   const float* __restrict__ X,
                                   float* __restrict__ Agg, int F) {
    int e = blockIdx.x;
    int c = threadIdx.x;
    int s = (int)src[e];
    int d = (int)dst[e];
    atomicAdd(&Agg[(size_t)d * F + c], X[(size_t)s * F + c]);
}

// ---------------------------------------------------------------------
// WMMA fp32 GEMM: Out[tm:tm+16, to:to+16] =
//   (Agg*inv) @ Wl^T + X @ Wr^T + bias     (K stepped by 4 per WMMA)
//
// A fragment (16x4, ISA layout): lane&15 = row M; VGPR0 = K{0 | 2},
// VGPR1 = K{1 | 3} selected by lane>>4.  B = W (row-major [O][K]) is
// exactly Wl^T in KxN form and loads with the identical pattern.
// C/D: VGPR r -> M = r (lanes 0-15) / r+8 (lanes 16-31), N = lane&15.
// ---------------------------------------------------------------------
__device__ __forceinline__ v2f ldfrag16x4(const float* __restrict__ base,
                                          int ld, int k, int lane) {
    const int r  = lane & 15;
    const int kk = k + ((lane >> 4) << 1);
    const float* p = base + (size_t)r * ld + kk;
    v2f f; f.x = p[0]; f.y = p[1];
    return f;
}

__global__ void __launch_bounds__(256)
sage_gemm_wmma(const float* __restrict__ Agg, const float* __restrict__ inv,
               const float* __restrict__ X,
               const float* __restrict__ Wl, const float* __restrict__ Wr,
               const float* __restrict__ bias, float* __restrict__ Out,
               int K, int O) {
    const int lane  = threadIdx.x & 31;
    const int wave  = threadIdx.x >> 5;
    const int tilesO = O >> 4;
    const int tile  = blockIdx.x * (blockDim.x >> 5) + wave;
    const int tm = (tile / tilesO) << 4;   // node-row base (exact: N % 16 == 0)
    const int to = (tile % tilesO) << 4;   // out-channel base

    const float* arow = Agg + (size_t)tm * K;
    const float* xrow = X   + (size_t)tm * K;
    const float* wlr  = Wl  + (size_t)to * K;
    const float* wrr  = Wr  + (size_t)to * K;
    const float  sA   = inv[tm + (lane & 15)];   // fused mean-division

    v8f acc = {};
    for (int k = 0; k < K; k += 4) {
        v2f a = ldfrag16x4(arow, K, k, lane);
        a.x *= sA; a.y *= sA;
        v2f b = ldfrag16x4(wlr, K, k, lane);
        acc = __builtin_amdgcn_wmma_f32_16x16x4_f32(false, a, false, b,
                                                    (short)0, acc, false, false);
    }
    for (int k = 0; k < K; k += 4) {
        v2f a = ldfrag16x4(xrow, K, k, lane);
        v2f b = ldfrag16x4(wrr, K, k, lane);
        acc = __builtin_amdgcn_wmma_f32_16x16x4_f32(false, a, false, b,
                                                    (short)0, acc, false, false);
    }

    const int nl = lane & 15;
    const int mb = (lane >> 4) << 3;
    const float bv = bias[to + nl];
    float* orow = Out + (size_t)(tm + mb) * O + to + nl;
#pragma unroll
    for (int r = 0; r < 8; ++r)
        orow[(size_t)r * O] = acc[r] + bv;
}

// ---------------------------------------------------------------------
// BatchNorm: column sums (thread == channel, coalesced) -> finalize ->
// fused affine+ReLU.
// ---------------------------------------------------------------------
__global__ void bn_stats_kernel(const float* __restrict__ H,
                                float* __restrict__ sums, int n, int F) {
    int c = threadIdx.x;                 // blockDim.x == F
    float s = 0.f, q = 0.f;
    for (int r = blockIdx.x; r < n; r += gridDim.x) {
        float v = H[(size_t)r * F + c];
        s += v; q += v * v;
    }
    atomicAdd(&sums[c], s);
    atomicAdd(&sums[F + c], q);
}

__global__ void bn_final_kernel(float* __restrict__ sums,
                                const float* __restrict__ gamma,
                                const float* __restrict__ beta,
                                int n, int F) {
    int c = threadIdx.x;                 // single block, blockDim.x == F
    if (c < F) {
        float mu  = sums[c] / (float)n;
        float var = sums[F + c] / (float)n - mu * mu;   // biased (torch BN train)
        float sc  = rsqrtf(var + BN_EPS) * gamma[c];
        sums[c]     = sc;                // scale
        sums[F + c] = beta[c] - mu * sc; // shift
    }
}

__global__ void bn_relu_kernel(float* __restrict__ H,
                               const float* __restrict__ prm,
                               size_t total, int Fmask_pow2) {
    size_t stride = (size_t)gridDim.x * blockDim.x;
    for (size_t i = (size_t)blockIdx.x * blockDim.x + threadIdx.x;
         i < total; i += stride) {
        int c = (int)(i & (size_t)(Fmask_pow2 - 1));   // F is a power of two
        float v = H[i] * prm[c] + prm[Fmask_pow2 + c];
        H[i] = fmaxf(v, 0.0f);
    }
}

// ---------------------------------------------------------------------
// Row-wise log-softmax, in place: one wave32 per row, 4 channels/lane.
// ---------------------------------------------------------------------
__global__ void logsoftmax_kernel(float* __restrict__ Out, int n, int F) {
    const int lane = threadIdx.x & 31;
    const int wave = threadIdx.x >> 5;
    const int row  = blockIdx.x * (blockDim.x >> 5) + wave;
    if (row >= n) return;
    float* p = Out + (size_t)row * F;
    const int per = F >> 5;              // 4 for F=128
    float v[4];
    float mx = -INFINITY;
#pragma unroll
    for (int i = 0; i < 4; ++i) {
        v[i] = (i < per) ? p[lane + (i << 5)] : -INFINITY;
        mx = fmaxf(mx, v[i]);
    }
#pragma unroll
    for (int o = 16; o > 0; o >>= 1) mx = fmaxf(mx, __shfl_xor(mx, o, 32));
    float s = 0.f;
#pragma unroll
    for (int i = 0; i < 4; ++i) if (i < per) s += expf(v[i] - mx);
#pragma unroll
    for (int o = 16; o > 0; o >>= 1) s += __shfl_xor(s, o, 32);
    float lse = logf(s);
#pragma unroll
    for (int i = 0; i < 4; ++i) if (i < per) p[lane + (i << 5)] = v[i] - mx - lse;
}

// ---------------------------------------------------------------------
extern "C" void kernel_launch(void* const* d_in, const int* in_sizes, int n_in,
                              void* d_out, int out_size, void* d_ws, size_t ws_size,
                              hipStream_t stream) {
    const float*     x    = (const float*)d_in[0];
    const long long* ei   = (const long long*)d_in[1];   // int64 (2,E)
    const long long* src  = ei;
    const long long* dst  = ei + EE;
    const float* Wl0 = (const float*)d_in[2];
    const float* bl0 = (const float*)d_in[3];
    const float* Wr0 = (const float*)d_in[4];
    const float* g0  = (const float*)d_in[5];
    const float* be0 = (const float*)d_in[6];
    const float* Wl1 = (const float*)d_in[7];
    const float* bl1 = (const float*)d_in[8];
    const float* Wr1 = (const float*)d_in[9];
    const float* g1  = (const float*)d_in[10];
    const float* be1 = (const float*)d_in[11];
    const float* Wl2 = (const float*)d_in[12];
    const float* bl2 = (const float*)d_in[13];
    const float* Wr2 = (const float*)d_in[14];
    float* out = (float*)d_out;

    // workspace carve-up (floats)
    float* ws  = (float*)d_ws;
    float* inv = ws;                              // NN
    float* agg = inv + NN;                        // NN*HID_C
    float* h0  = agg + (size_t)NN * HID_C;        // NN*HID_C
    float* h1  = h0  + (size_t)NN * HID_C;        // NN*HID_C
    float* bn  = h1  + (size_t)NN * HID_C;        // 2*HID_C (sums -> scale/shift)

    const size_t elemsH = (size_t)NN * HID_C;
    const int tb = 256;

    // GEMM grids: tiles = (N/16)*(O/16); 8 waves (tiles) per 256-thread block
    const int blk_g_hid = (NN / 16) * (HID_C / 16) / 8;   // 6250
    const int blk_g_out = (NN / 16) * (OUT_C / 16) / 8;   // 3125

    // ---- mean denominators (shared by all layers) ----
    hipMemsetAsync(inv, 0, NN * sizeof(float), stream);
    count_kernel<<<(EE + tb - 1) / tb, tb, 0, stream>>>(dst, inv, EE);
    inv_kernel<<<(NN + tb - 1) / tb, tb, 0, stream>>>(inv, NN);

    // ---- layer 0: SAGE(x) -> h0 ; BN+ReLU ----
    hipMemsetAsync(agg, 0, (size_t)NN * IN_C * sizeof(float), stream);
    scatter_add_kernel<<<EE, IN_C, 0, stream>>>(src, dst, x, agg, IN_C);
    sage_gemm_wmma<<<blk_g_hid, tb, 0, stream>>>(agg, inv, x, Wl0, Wr0, bl0, h0,
                                                 IN_C, HID_C);
    hipMemsetAsync(bn, 0, 2 * HID_C * sizeof(float), stream);
    bn_stats_kernel<<<512, HID_C, 0, stream>>>(h0, bn, NN, HID_C);
    bn_final_kernel<<<1, HID_C, 0, stream>>>(bn, g0, be0, NN, HID_C);
    bn_relu_kernel<<<4096, tb, 0, stream>>>(h0, bn, elemsH, HID_C);

    // ---- layer 1: SAGE(h0) -> h1 ; BN+ReLU ----
    hipMemsetAsync(agg, 0, (size_t)NN * HID_C * sizeof(float), stream);
    scatter_add_kernel<<<EE, HID_C, 0, stream>>>(src, dst, h0, agg, HID_C);
    sage_gemm_wmma<<<blk_g_hid, tb, 0, stream>>>(agg, inv, h0, Wl1, Wr1, bl1, h1,
                                                 HID_C, HID_C);
    hipMemsetAsync(bn, 0, 2 * HID_C * sizeof(float), stream);
    bn_stats_kernel<<<512, HID_C, 0, stream>>>(h1, bn, NN, HID_C);
    bn_final_kernel<<<1, HID_C, 0, stream>>>(bn, g1, be1, NN, HID_C);
    bn_relu_kernel<<<4096, tb, 0, stream>>>(h1, bn, elemsH, HID_C);

    // ---- layer 2: SAGE(h1) -> out ; log_softmax in place ----
    hipMemsetAsync(agg, 0, (size_t)NN * HID_C * sizeof(float), stream);
    scatter_add_kernel<<<EE, HID_C, 0, stream>>>(src, dst, h1, agg, HID_C);
    sage_gemm_wmma<<<blk_g_out, tb, 0, stream>>>(agg, inv, h1, Wl2, Wr2, bl2, out,
                                                 HID_C, OUT_C);
    logsoftmax_kernel<<<(NN + 7) / 8, tb, 0, stream>>>(out, NN, OUT_C);
}